// Deformable_Batchformer_31061203485392
// MI455X (gfx1250) — compile-verified
//
#include <hip/hip_runtime.h>
#include <math.h>

// ---------------------------------------------------------------------------
// Deformable Batchformer for MI455X (gfx1250, wave32).
// GEMM stages: V_WMMA_F32_16X16X4_F32, 16x64 tile/wave (4 acc chains),
// weight tile staged to LDS with global_load_async_to_lds_b128 (ASYNCcnt).
// ---------------------------------------------------------------------------

typedef float v2f __attribute__((ext_vector_type(2)));
typedef float v8f __attribute__((ext_vector_type(8)));

#define BN   6
#define CCH  128
#define HID  64
#define HH   80
#define WW   80
#define HWP  (HH*WW)        // 6400
#define BB   (BN*BN)        // 36
#define CHW  (CCH*HWP)      // 819200
#define TOK  (BN*HWP)       // 38400
#define NN4  (4*HWP)        // 25600

// ---------------------------------------------------------------------------
// fp32 WMMA GEMM, K = 128 fixed:  Y[o,p] = sum_k Wm[o,k]*X[k,p] (+bias,+R,relu)
// Compile-time layouts. Block = 128 thr = 4 waves; wave tile = 16(O) x 64(P),
// block tile = 16 x 256. Weight tile (16x128, 8KB) async-staged into LDS.
// No divergence anywhere -> EXEC all ones for every WMMA (ISA requirement).
// ---------------------------------------------------------------------------
template<int SXC, int SXP, int SYO, int SYP, int RELU, int HASRES>
__global__ __launch_bounds__(128)
void gemm_wmma_f32_kernel(const float* __restrict__ Wm,
                          const float* __restrict__ X,
                          const float* __restrict__ bias,
                          const float* __restrict__ R,
                          float* __restrict__ Y,
                          long long xbs, long long ybs)
{
    __shared__ float sW[16 * 128];          // one 16x128 weight tile

    const int o0 = blockIdx.y * 16;

    // ---- async DMA of weight tile into LDS (gfx1250 ASYNC path) ----------
    {
        const unsigned lds0 = (unsigned)(unsigned long long)(const void*)sW;
        #pragma unroll
        for (int p = 0; p < 4; ++p) {
            const int e4  = p * 128 + (int)threadIdx.x;   // float4 index 0..511
            const int row = e4 >> 5;                      // 0..15
            const int c4  = e4 & 31;                      // 0..31
            const float* g  = Wm + ((long long)(o0 + row) << 7) + (c4 << 2);
            const unsigned l = lds0 + (unsigned)(e4 << 4);
            asm volatile("global_load_async_to_lds_b128 %0, %1, off"
                         :: "v"(l), "v"(g) : "memory");
        }
        asm volatile("s_wait_asynccnt 0x0" ::: "memory");
        __syncthreads();
    }

    const int wave = threadIdx.x >> 5;
    const int lane = threadIdx.x & 31;
    const int half = lane >> 4;            // 0: lanes 0-15, 1: lanes 16-31
    const int l16  = lane & 15;

    const float* Xb = X + (long long)blockIdx.z * xbs;
    float*       Yb = Y + (long long)blockIdx.z * ybs;

    const int pw = (blockIdx.x * 4 + wave) * 64;   // this wave's 64-wide strip

    long long col[4];
    #pragma unroll
    for (int t = 0; t < 4; ++t)
        col[t] = (long long)(pw + t*16 + l16) * SXP;

    const float* aRow = &sW[l16 * 128];    // ds_load_b64 A-fragments

    v8f acc[4] = {v8f{}, v8f{}, v8f{}, v8f{}};

    #pragma unroll 4
    for (int k = 0; k < 128; k += 4) {
        const int k2 = k + 2*half;
        v2f a;
        {   const float2 av = *(const float2*)&aRow[k2];
            a.x = av.x; a.y = av.y; }

        v2f b[4];
        if (SXC == 1) {
            // token-major: K contiguous -> float2 loads
            #pragma unroll
            for (int t = 0; t < 4; ++t) {
                const float2 bv = *(const float2*)&Xb[col[t] + k2];
                b[t].x = bv.x; b[t].y = bv.y;
            }
        } else {
            // conv-major: K strided by HW
            #pragma unroll
            for (int t = 0; t < 4; ++t) {
                b[t].x = Xb[(long long)(k2    ) * SXC + col[t]];
                b[t].y = Xb[(long long)(k2 + 1) * SXC + col[t]];
            }
            // branch-free prefetch of next K rows (clamped, stays in panel)
            const int kp = (k2 + 4 < 124) ? (k2 + 4) : 123;
            __builtin_prefetch(&Xb[(long long)kp * SXC + col[0]], 0, 3);
        }

        #pragma unroll
        for (int t = 0; t < 4; ++t)
            acc[t] = __builtin_amdgcn_wmma_f32_16x16x4_f32(
                         false, a, false, b[t], (short)0, acc[t], false, false);
    }

    // ---- epilogue: C/D layout -> vgpr v holds row m = v + 8*half, col l16 --
    #pragma unroll
    for (int t = 0; t < 4; ++t) {
        const int cidx = pw + t*16 + l16;
        if (SYO == 1) {
            // token-major out: rows contiguous -> two float4 stores
            float* dst = Yb + (long long)cidx * SYP + o0 + 8*half;
            const float* rsd = HASRES ? (R + (long long)cidx * SYP + o0 + 8*half)
                                      : dst;
            float vals[8];
            #pragma unroll
            for (int v = 0; v < 8; ++v) {
                float q = acc[t][v] + bias[o0 + 8*half + v];
                if (HASRES) q += rsd[v];
                if (RELU)   q = fmaxf(q, 0.0f);
                vals[v] = q;
            }
            *(float4*)(dst)     = make_float4(vals[0], vals[1], vals[2], vals[3]);
            *(float4*)(dst + 4) = make_float4(vals[4], vals[5], vals[6], vals[7]);
        } else {
            #pragma unroll
            for (int v = 0; v < 8; ++v) {
                const int m = v + 8*half;
                float q = acc[t][v] + bias[o0 + m];
                if (RELU) q = fmaxf(q, 0.0f);
                Yb[(long long)(o0 + m) * SYO + (long long)cidx * SYP] = q;
            }
        }
    }
}

// ---------------------------------------------------------------------------
// 9x9 grouped conv (2 in-ch per group) + exact GELU. One (n, out-ch) pair per
// block -> the 162 taps live in LDS; interior pixels take unguarded fast path.
// ---------------------------------------------------------------------------
__global__ __launch_bounds__(256)
void dwconv9x9_gelu_kernel(const float* __restrict__ xf,
                           const float* __restrict__ Wd1,
                           float* __restrict__ z)
{
    __shared__ float w[162];
    const int no = blockIdx.y;             // n*64 + o
    const int o  = no & 63;
    const int n  = no >> 6;
    if (threadIdx.x < 162)
        w[threadIdx.x] = Wd1[(long long)o * 162 + threadIdx.x];
    __syncthreads();

    const int pix = blockIdx.x * 256 + threadIdx.x;   // 25*256 == HWP exactly
    const int i = n / BN, j = n % BN;
    const int py = pix / WW, px = pix % WW;
    const int src = (o < 32) ? i : j;
    const int ch0 = (2 * o) & 63;

    const float* img0 = xf + ((long long)src * HID + ch0) * HWP;
    const float* img1 = img0 + HWP;

    float acc = 0.0f;
    if (py >= 4 && py < HH-4 && px >= 4 && px < WW-4) {
        #pragma unroll
        for (int dy = 0; dy < 9; ++dy) {
            const int base = (py + dy - 4) * WW + (px - 4);
            #pragma unroll
            for (int dx = 0; dx < 9; ++dx)
                acc += img0[base + dx] * w[dy*9 + dx]
                     + img1[base + dx] * w[81 + dy*9 + dx];
        }
    } else {
        for (int dy = 0; dy < 9; ++dy) {
            const int yy = py + dy - 4;
            if (yy < 0 || yy >= HH) continue;
            for (int dx = 0; dx < 9; ++dx) {
                const int xx = px + dx - 4;
                if (xx < 0 || xx >= WW) continue;
                acc += img0[yy*WW + xx] * w[dy*9 + dx]
                     + img1[yy*WW + xx] * w[81 + dy*9 + dx];
            }
        }
    }
    z[(long long)no * HWP + pix] =
        0.5f * acc * (1.0f + erff(acc * 0.70710678118654752f));
}

// ---------------------------------------------------------------------------
// flow = tanh(Wd2 @ z) * mask (diagonal pairs zeroed)
// ---------------------------------------------------------------------------
__global__ void flow_kernel(const float* __restrict__ z,
                            const float* __restrict__ Wd2,
                            float* __restrict__ flow)
{
    const int idx = blockIdx.x * blockDim.x + threadIdx.x;
    if (idx >= BB * HWP) return;
    const int pix = idx % HWP;
    const int n   = idx / HWP;
    const float* zp = z + (long long)n * HID * HWP + pix;
    float a0 = 0.0f, a1 = 0.0f;
    for (int c = 0; c < HID; ++c) {
        const float v = zp[(long long)c * HWP];
        a0 += v * Wd2[c];
        a1 += v * Wd2[HID + c];
    }
    const float m = ((n / BN) == (n % BN)) ? 0.0f : 1.0f;
    flow[(long long)n * 2 * HWP + pix]       = tanhf(a0) * m;
    flow[(long long)n * 2 * HWP + HWP + pix] = tanhf(a1) * m;
}

// ---------------------------------------------------------------------------
// Bilinear flow-warp of x[n%6] -> kv (written into d_out kv_d slice).
// ---------------------------------------------------------------------------
__global__ void warp_kernel(const float* __restrict__ x,
                            const float* __restrict__ flow,
                            float* __restrict__ kv)
{
    const int idx = blockIdx.x * blockDim.x + threadIdx.x;
    if (idx >= BB * HWP) return;
    const int pix = idx % HWP;
    const int n   = idx / HWP;
    const int py = pix / WW, px = pix % WW;

    const float f0 = flow[(long long)n * 2 * HWP + pix];
    const float f1 = flow[(long long)n * 2 * HWP + HWP + pix];
    const float gy = (-1.0f + 2.0f * py / (float)(HH - 1)) + f0 * (9.0f / WW);
    const float gx = (-1.0f + 2.0f * px / (float)(WW - 1)) + f1 * (9.0f / HH);
    const float ix = (gx + 1.0f) * 0.5f * (WW - 1);
    const float iy = (gy + 1.0f) * 0.5f * (HH - 1);

    const float y0f = floorf(iy), x0f = floorf(ix);
    const float wy = iy - y0f, wx = ix - x0f;
    const int y0 = (int)y0f, x0 = (int)x0f;
    const int y1 = y0 + 1,   x1 = x0 + 1;

    const float ok00 = (y0 >= 0 && y0 < HH && x0 >= 0 && x0 < WW) ? 1.f : 0.f;
    const float ok01 = (y0 >= 0 && y0 < HH && x1 >= 0 && x1 < WW) ? 1.f : 0.f;
    const float ok10 = (y1 >= 0 && y1 < HH && x0 >= 0 && x0 < WW) ? 1.f : 0.f;
    const float ok11 = (y1 >= 0 && y1 < HH && x1 >= 0 && x1 < WW) ? 1.f : 0.f;
    const int y0c = min(max(y0, 0), HH-1), x0c = min(max(x0, 0), WW-1);
    const int y1c = min(max(y1, 0), HH-1), x1c = min(max(x1, 0), WW-1);

    const float w00 = (1.f-wy)*(1.f-wx)*ok00, w01 = (1.f-wy)*wx*ok01;
    const float w10 = wy*(1.f-wx)*ok10,       w11 = wy*wx*ok11;
    const long long o00 = y0c*WW + x0c, o01 = y0c*WW + x1c;
    const long long o10 = y1c*WW + x0c, o11 = y1c*WW + x1c;

    const float* img = x + (long long)(n % BN) * CHW;
    float* dst = kv + (long long)n * CHW + pix;
    for (int c = 0; c < CCH; ++c) {
        const float* ic = img + (long long)c * HWP;
        dst[(long long)c * HWP] = w00*ic[o00] + w01*ic[o01] + w10*ic[o10] + w11*ic[o11];
    }
}

// ---------------------------------------------------------------------------
// Cross-batch attention (softmax width 6, head-dim 32) + residual -> y1.
// ---------------------------------------------------------------------------
__global__ void attn_kernel(const float* __restrict__ x,
                            const float* __restrict__ kk,
                            const float* __restrict__ vv,
                            float* __restrict__ y1)
{
    const int idx = blockIdx.x * blockDim.x + threadIdx.x;
    if (idx >= BN * NN4) return;
    const int ni = idx % NN4;
    const int k  = idx / NN4;
    const float* qb = x + (long long)k * CHW;    // viewed (32, NN4)

    float e[BN];
    for (int j = 0; j < BN; ++j) {
        const float* kb = kk + (long long)(k*BN + j) * CHW;
        float acc = 0.0f;
        for (int si = 0; si < 32; ++si)
            acc += qb[(long long)si * NN4 + ni] * kb[(long long)si * NN4 + ni];
        e[j] = acc * 0.17677669529663689f;        // 32^-0.5
    }
    float mx = e[0];
    for (int j = 1; j < BN; ++j) mx = fmaxf(mx, e[j]);
    float s = 0.0f;
    for (int j = 0; j < BN; ++j) { e[j] = expf(e[j] - mx); s += e[j]; }
    const float inv = 1.0f / s;

    const int pix = ni % HWP, chb = ni / HWP;
    const long long token = (long long)k * HWP + pix;
    for (int si = 0; si < 32; ++si) {
        float acc = 0.0f;
        for (int j = 0; j < BN; ++j)
            acc += e[j] * inv * vv[(long long)(k*BN + j) * CHW + (long long)si * NN4 + ni];
        y1[token * CCH + (si*4 + chb)] = acc + qb[(long long)si * NN4 + ni];
    }
}

// ---------------------------------------------------------------------------
// LayerNorm(128): one wave32 per token, 4 ch/lane, __shfl_xor reductions.
// ---------------------------------------------------------------------------
__global__ void ln_kernel(const float* __restrict__ in,
                          const float* __restrict__ g,
                          const float* __restrict__ be,
                          float* __restrict__ out, int transposeOut)
{
    const int lane = threadIdx.x & 31;
    const long long token = (long long)blockIdx.x * 4 + (threadIdx.x >> 5);
    const float4 v = ((const float4*)(in + token * CCH))[lane];

    float sum = v.x + v.y + v.z + v.w;
    for (int off = 16; off > 0; off >>= 1) sum += __shfl_xor(sum, off, 32);
    const float mean = sum * (1.0f / CCH);
    const float dx = v.x-mean, dy = v.y-mean, dz = v.z-mean, dw = v.w-mean;
    float vs = dx*dx + dy*dy + dz*dz + dw*dw;
    for (int off = 16; off > 0; off >>= 1) vs += __shfl_xor(vs, off, 32);
    const float rstd = rsqrtf(vs * (1.0f / CCH) + 1e-5f);

    const int c0 = lane * 4;
    const float r0 = dx*rstd*g[c0+0] + be[c0+0];
    const float r1 = dy*rstd*g[c0+1] + be[c0+1];
    const float r2 = dz*rstd*g[c0+2] + be[c0+2];
    const float r3 = dw*rstd*g[c0+3] + be[c0+3];

    if (!transposeOut) {
        ((float4*)(out + token * CCH))[lane] = make_float4(r0, r1, r2, r3);
    } else {
        const int kimg = (int)(token / HWP), pix = (int)(token % HWP);
        float* ob = out + (long long)kimg * CHW + pix;
        ob[(long long)(c0+0) * HWP] = r0;
        ob[(long long)(c0+1) * HWP] = r1;
        ob[(long long)(c0+2) * HWP] = r2;
        ob[(long long)(c0+3) * HWP] = r3;
    }
}

// ---------------------------------------------------------------------------
extern "C" void kernel_launch(void* const* d_in, const int* in_sizes, int n_in,
                              void* d_out, int out_size, void* d_ws, size_t ws_size,
                              hipStream_t stream)
{
    const float* x   = (const float*)d_in[0];
    const float* Wf  = (const float*)d_in[1];
    const float* bf  = (const float*)d_in[2];
    const float* Wk  = (const float*)d_in[3];
    const float* bk  = (const float*)d_in[4];
    const float* Wv  = (const float*)d_in[5];
    const float* bv  = (const float*)d_in[6];
    const float* Wd1 = (const float*)d_in[7];
    const float* Wd2 = (const float*)d_in[8];
    const float* W1  = (const float*)d_in[9];
    const float* b1  = (const float*)d_in[10];
    const float* W2  = (const float*)d_in[11];
    const float* b2  = (const float*)d_in[12];
    const float* g1  = (const float*)d_in[13];
    const float* be1 = (const float*)d_in[14];
    const float* g2  = (const float*)d_in[15];
    const float* be2 = (const float*)d_in[16];

    float* out  = (float*)d_out;
    float* outY = out;                                    // (6,128,80,80)
    float* kvD  = out + (long long)BN * CHW;              // (36,128,80,80) == kv
    float* outX = kvD + (long long)BB * CHW;              // copy of x

    // workspace layout (floats) with dead-buffer reuse
    float* ws   = (float*)d_ws;
    float* xf   = ws;                                     //  2,457,600 (6,64,HW)
    float* z    = ws + 2457600LL;                         // 14,745,600 (36,64,HW)
    float* y1   = z;                                      // reuse z (post-flow)
    float* yln  = z + 4915200LL;                          // reuse z
    float* flw  = ws + 17203200LL;                        //    460,800 (36,2,HW)
    float* kk   = ws + 17664000LL;                        // 29,491,200 (36,128,HW)
    float* ff   = kk;                                     // reuse kk (post-attn)
    float* y2   = kk + 4915200LL;                         // reuse kk
    float* vv   = ws + 47155200LL;                        // 29,491,200 (36,128,HW)

    dim3 blk(128);

    // 1) x_ = Wf @ x + bf            (6 batches, O=64, K=128, P=6400)
    gemm_wmma_f32_kernel<HWP,1,HWP,1,0,0>
        <<<dim3(HWP/256, HID/16, BN), blk, 0, stream>>>(
        Wf, x, bf, nullptr, xf, (long long)CHW, (long long)HID*HWP);

    // 2) grouped 9x9 conv + gelu -> z   (one (n,o) pair per block)
    dwconv9x9_gelu_kernel<<<dim3(HWP/256, BB*HID), 256, 0, stream>>>(xf, Wd1, z);

    // 3) flow = tanh(Wd2 @ z) * mask
    flow_kernel<<<(BB*HWP + 255) / 256, 256, 0, stream>>>(z, Wd2, flw);

    // 4) kv = flow_warp(x[j], offset) -> written straight into kv_d output
    warp_kernel<<<(BB*HWP + 255) / 256, 256, 0, stream>>>(x, flw, kvD);

    // 5) kk = Wk @ kv + bk ; 6) vv = Wv @ kv + bv  (36 batches, 128x128xHW)
    gemm_wmma_f32_kernel<HWP,1,HWP,1,0,0>
        <<<dim3(HWP/256, CCH/16, BB), blk, 0, stream>>>(
        Wk, kvD, bk, nullptr, kk, (long long)CHW, (long long)CHW);
    gemm_wmma_f32_kernel<HWP,1,HWP,1,0,0>
        <<<dim3(HWP/256, CCH/16, BB), blk, 0, stream>>>(
        Wv, kvD, bv, nullptr, vv, (long long)CHW, (long long)CHW);

    // 7) attention + residual -> y1 (token-major)
    attn_kernel<<<(BN*NN4 + 255) / 256, 256, 0, stream>>>(x, kk, vv, y1);

    // 8) LN1 -> yln
    ln_kernel<<<TOK/4, 128, 0, stream>>>(y1, g1, be1, yln, 0);

    // 9) ff = relu(W1 @ yln^T + b1)  (O=128, K=128, P=38400 tokens)
    gemm_wmma_f32_kernel<1,CCH,1,CCH,1,0>
        <<<dim3(TOK/256, CCH/16, 1), blk, 0, stream>>>(
        W1, yln, b1, nullptr, ff, 0, 0);

    // 10) y2 = W2 @ ff^T + b2 + yln (residual)
    gemm_wmma_f32_kernel<1,CCH,1,CCH,0,1>
        <<<dim3(TOK/256, CCH/16, 1), blk, 0, stream>>>(
        W2, ff, b2, yln, y2, 0, 0);

    // 11) LN2 -> final y, transposed to (6,128,80,80)
    ln_kernel<<<TOK/4, 128, 0, stream>>>(y2, g2, be2, outY, 1);

    // 12) third output = x (stop_gradient passthrough)
    hipMemcpyAsync(outX, x, (size_t)BN * CHW * sizeof(float),
                   hipMemcpyDeviceToDevice, stream);
}